// WeightedAverage_55551107006568
// MI455X (gfx1250) — compile-verified
//
#include <hip/hip_runtime.h>
#include <hip/hip_bf16.h>
#include <stdint.h>

// 3x3 softmax-weighted local average, simplified:
//   out = sum_k l_k * exp(m - l_k) / sum_k exp(m - l_k),  m = min_k l_k
// (the x^2 term of the reference cancels identically in the softmax,
//  including under jax's max-subtraction: weights are exp(min_j l_j - l_k))

#define TW 64              // tile width (output cols per block)
#define TH 32              // tile height (output rows per block)
#define RPT 8              // rows per thread (TH / blockDim.y)
#define HALO_W (TW + 2)    // 66
#define HALO_H (TH + 2)    // 34
#define TILE_N (HALO_W * HALO_H)  // 2244 floats = 8976 B LDS
#define NTHREADS 256

__global__ __launch_bounds__(NTHREADS)
void WeightedAverage_stencil_kernel(const float* __restrict__ src,
                                    float* __restrict__ dst,
                                    int H, int W)
{
    __shared__ float tile[TILE_N];

    const int tx  = threadIdx.x;                 // 0..63
    const int ty  = threadIdx.y;                 // 0..3
    const int tid = tx + ty * TW;                // 0..255

    const int gx0 = blockIdx.x * TW - 1;         // global col of tile col 0 (halo)
    const int gy0 = blockIdx.y * TH - 1;         // global row of tile row 0 (halo)
    const size_t imgbase = (size_t)blockIdx.z * (size_t)H * (size_t)W;
    const float* __restrict__ img = src + imgbase;

    // LDS byte offset of the tile base: flat shared addresses carry the
    // wave-relative LDS offset in their low 32 bits (aperture layout).
    const unsigned lbase = (unsigned)(uintptr_t)(&tile[0]);

    // ---- stage halo tile into LDS via CDNA5 async global->LDS copies ----
    #pragma unroll
    for (int k = 0; k < (TILE_N + NTHREADS - 1) / NTHREADS; ++k) {
        const int li = tid + k * NTHREADS;
        if (li < TILE_N) {
            const int r  = li / HALO_W;
            const int c  = li - r * HALO_W;
            const int gy = gy0 + r;
            const int gx = gx0 + c;
            const unsigned loff = lbase + (unsigned)li * 4u;
            if ((unsigned)gy < (unsigned)H && (unsigned)gx < (unsigned)W) {
                const float* gp = img + ((size_t)gy * (size_t)W + (size_t)gx);
                // vdst = LDS byte address, vaddr = 64-bit global address
                asm volatile("global_load_async_to_lds_b32 %0, %1, off"
                             :: "v"(loff), "v"(gp)
                             : "memory");
            } else {
                tile[li] = 0.0f;   // zero padding (boundary blocks only)
            }
        }
    }
    // wait for this wave's async transfers, then workgroup barrier
    asm volatile("s_wait_asynccnt 0x0" ::: "memory");
    __syncthreads();

    // ---- compute: rolling 3x3 register window down the column ----
    const float LOG2E = 1.44269504088896340f;
    const float LN2   = 0.69314718055994531f;

    const int tr = ty * RPT;                     // first output row in tile coords
    const float* __restrict__ trow = &tile[tr * HALO_W + tx];

    // values pre-scaled by log2(e): weights are exp2(m_s - t_k)
    float a0 = trow[0] * LOG2E, a1 = trow[1] * LOG2E, a2 = trow[2] * LOG2E;
    trow += HALO_W;
    float b0 = trow[0] * LOG2E, b1 = trow[1] * LOG2E, b2 = trow[2] * LOG2E;

    float* __restrict__ outp = dst + imgbase
        + (size_t)(blockIdx.y * TH + tr) * (size_t)W
        + (size_t)(blockIdx.x * TW + tx);

    #pragma unroll
    for (int o = 0; o < RPT; ++o) {
        trow += HALO_W;
        const float c0 = trow[0] * LOG2E;
        const float c1 = trow[1] * LOG2E;
        const float c2 = trow[2] * LOG2E;

        const float m =
            __builtin_fminf(
                __builtin_fminf(__builtin_fminf(a0, a1), __builtin_fminf(a2, b0)),
                __builtin_fminf(__builtin_fminf(b1, b2),
                                __builtin_fminf(__builtin_fminf(c0, c1), c2)));

        const float w0 = __builtin_amdgcn_exp2f(m - a0);
        const float w1 = __builtin_amdgcn_exp2f(m - a1);
        const float w2 = __builtin_amdgcn_exp2f(m - a2);
        const float w3 = __builtin_amdgcn_exp2f(m - b0);
        const float w4 = __builtin_amdgcn_exp2f(m - b1);
        const float w5 = __builtin_amdgcn_exp2f(m - b2);
        const float w6 = __builtin_amdgcn_exp2f(m - c0);
        const float w7 = __builtin_amdgcn_exp2f(m - c1);
        const float w8 = __builtin_amdgcn_exp2f(m - c2);

        const float den = (((w0 + w1) + (w2 + w3)) + ((w4 + w5) + (w6 + w7))) + w8;

        float num = a0 * w0;
        num = __builtin_fmaf(a1, w1, num);
        num = __builtin_fmaf(a2, w2, num);
        num = __builtin_fmaf(b0, w3, num);
        num = __builtin_fmaf(b1, w4, num);
        num = __builtin_fmaf(b2, w5, num);
        num = __builtin_fmaf(c0, w6, num);
        num = __builtin_fmaf(c1, w7, num);
        num = __builtin_fmaf(c2, w8, num);

        // sum l_k w_k = ln2 * sum t_k w_k  (t = l * log2e)
        // Non-temporal store: output is write-once; keep the (L2-resident)
        // input from being evicted by the output stream.
        __builtin_nontemporal_store(num * (LN2 * __builtin_amdgcn_rcpf(den)), outp);
        outp += W;

        a0 = b0; a1 = b1; a2 = b2;
        b0 = c0; b1 = c1; b2 = c2;
    }
}

extern "C" void kernel_launch(void* const* d_in, const int* in_sizes, int n_in,
                              void* d_out, int out_size, void* d_ws, size_t ws_size,
                              hipStream_t stream) {
    (void)n_in; (void)out_size; (void)d_ws; (void)ws_size;
    const float* x = (const float*)d_in[0];
    float* out = (float*)d_out;

    const int H = 1536, W = 1536;
    const int N = in_sizes[0] / (H * W);   // 16

    dim3 block(TW, NTHREADS / TW, 1);      // (64, 4)
    dim3 grid(W / TW, H / TH, N);          // (24, 48, 16)
    hipLaunchKernelGGL(WeightedAverage_stencil_kernel, grid, block, 0, stream,
                       x, out, H, W);
}